// TokenClassificationModelCRF_59055800320228
// MI455X (gfx1250) — compile-verified
//
#include <hip/hip_runtime.h>
#include <hip/hip_bf16.h>

typedef __attribute__((ext_vector_type(2))) float v2f;
typedef __attribute__((ext_vector_type(8))) float v8f;

#define Bv 32
#define Sv 2048
#define Hv 1024
#define Lv 13
#define NCH 64           // chunks of the sequence scan
#define CHUNK 32         // steps per chunk (64*32 >= 2047)
#define WROWS 14         // 13 rows of W + one zero pad row
#define WSTRIDE 1026     // pad stride to avoid 64-bank conflicts
#define NEG_INF (-1e30f)

// ---------------------------------------------------------------------------
// Kernel 1: logits = hidden @ W^T + b  via V_WMMA_F32_16X16X4_F32
// One wave computes a 16x16 (rows x classes) fp32 tile; K-loop over H=1024.
// W is staged in LDS once per block (bank-conflict-padded), B-frags come from
// ds_load_b64; A-frags are 8B global loads per lane. Bandwidth-bound by design.
// ---------------------------------------------------------------------------
__global__ __launch_bounds__(256) void gemm_logits_kernel(
    const float* __restrict__ hidden, const float* __restrict__ W,
    const float* __restrict__ bias, float* __restrict__ logits)
{
  __shared__ float wlds[WROWS * WSTRIDE];
  const int tid = threadIdx.x;
  for (int idx = tid; idx < WROWS * Hv; idx += 256) {
    int r = idx >> 10, c = idx & (Hv - 1);
    wlds[r * WSTRIDE + c] = (r < Lv) ? W[r * Hv + c] : 0.0f;
  }
  __syncthreads();

  const int lane = tid & 31;
  const int wave = tid >> 5;
  const int gw   = blockIdx.x * 8 + wave;   // 4096 waves cover 65536 rows
  const int base = gw * 16;
  const int mrow = lane & 15;               // A frag: M = lane&15
  const int koff = (lane >> 4) * 2;         // A frag: VGPR0/1 hold K=koff,koff+1
  const int n    = lane & 15;               // B frag: N = lane&15 (mirror layout)
  const int nld  = (n < Lv) ? n : Lv;       // pad classes 13..15 -> zero row

  const float* ap = hidden + (size_t)(base + mrow) * Hv + koff;
  const float* bp = &wlds[nld * WSTRIDE + koff];

  v8f acc = {0.f, 0.f, 0.f, 0.f, 0.f, 0.f, 0.f, 0.f};
#pragma unroll 8
  for (int k = 0; k < Hv; k += 4) {
    v2f a = *(const v2f*)(ap + k);          // global_load_b64
    v2f b = *(const v2f*)(bp + k);          // ds_load_b64
    acc = __builtin_amdgcn_wmma_f32_16x16x4_f32(
        /*neg_a=*/false, a, /*neg_b=*/false, b,
        /*c_mod=*/(short)0, acc, /*reuse_a=*/false, /*reuse_b=*/false);
  }

  const float bn = (n < Lv) ? bias[n] : 0.f;
  const int mhi = (lane >> 4) * 8;          // D frag: lanes>=16 hold M=r+8
  if (n < Lv) {
#pragma unroll
    for (int r = 0; r < 8; ++r) {
      int row = base + r + mhi;
      logits[(size_t)row * Lv + n] = acc[r] + bn;
    }
  }
}

// ---------------------------------------------------------------------------
// Kernel 2: chunked parallel scan of the CRF forward algorithm.
// Each block = one (batch, chunk); propagates a 13x16 log-semiring transfer
// matrix through CHUNK steps. cur lives in LDS; thread (i,j) does one lse.
// ---------------------------------------------------------------------------
__global__ __launch_bounds__(224) void crf_scan_kernel(
    const float* __restrict__ logits, const int* __restrict__ mask,
    const float* __restrict__ trans, float* __restrict__ mats)
{
  __shared__ float cur[13 * 16];
  const int b = blockIdx.x / NCH;
  const int c = blockIdx.x % NCH;
  const int t = threadIdx.x;
  const int i = t >> 4, j = t & 15;
  const bool active = (t < 208) && (j < Lv);

  float Tcol[Lv];
  if (active) {
#pragma unroll
    for (int k = 0; k < Lv; ++k) Tcol[k] = trans[k * Lv + j];
  }
  if (t < 208) cur[t] = (i == j) ? 0.f : NEG_INF;   // identity in log space
  __syncthreads();

  for (int step = 0; step < CHUNK; ++step) {
    const int s = 1 + c * CHUNK + step;             // steps 1..2047
    const bool act = active && (s < Sv);
    float val = 0.f;
    if (act) {
      float v[Lv];
      float mx = NEG_INF;
#pragma unroll
      for (int k = 0; k < Lv; ++k) {
        v[k] = cur[i * 16 + k] + Tcol[k];
        mx = fmaxf(mx, v[k]);
      }
      float sum = 0.f;
#pragma unroll
      for (int k = 0; k < Lv; ++k) sum += __expf(v[k] - mx);
      val = mx + __logf(sum) + logits[((size_t)b * Sv + s) * Lv + j];
      if (mask[b * Sv + s] == 0) val = cur[i * 16 + j];  // masked step = identity
    }
    __syncthreads();
    if (act) cur[i * 16 + j] = val;
    __syncthreads();
  }
  if (t < 208) mats[(size_t)blockIdx.x * 208 + t] = cur[t];
}

// ---------------------------------------------------------------------------
// Kernel 3: fold start vector through the 64 chunk matrices; add end terms.
// One wave per batch; lane j owns state j; v broadcast via __shfl.
// ---------------------------------------------------------------------------
__global__ __launch_bounds__(32) void crf_combine_kernel(
    const float* __restrict__ logits, const float* __restrict__ mats,
    const float* __restrict__ startt, const float* __restrict__ endt,
    float* __restrict__ den)
{
  const int b = blockIdx.x;
  const int j = threadIdx.x;
  const int jj = (j < Lv) ? j : 0;
  float v = (j < Lv) ? startt[j] + logits[(size_t)b * Sv * Lv + j] : NEG_INF;

  for (int c = 0; c < NCH; ++c) {
    const float* M = mats + ((size_t)b * NCH + c) * 208;
    float tv[Lv];
    float mx = NEG_INF;
#pragma unroll
    for (int i = 0; i < Lv; ++i) {
      float sv = __shfl(v, i);               // uniform for all 32 lanes
      tv[i] = sv + M[i * 16 + jj];
      mx = fmaxf(mx, tv[i]);
    }
    float sum = 0.f;
#pragma unroll
    for (int i = 0; i < Lv; ++i) sum += __expf(tv[i] - mx);
    float nv = mx + __logf(sum);
    v = (j < Lv) ? nv : NEG_INF;
  }

  float w = (j < Lv) ? v + endt[j] : NEG_INF;
  float mx2 = w;
  for (int o = 16; o; o >>= 1) mx2 = fmaxf(mx2, __shfl_xor(mx2, o));
  float e = __expf(w - mx2);                  // lanes >= 13 contribute exp(-inf)=0
  for (int o = 16; o; o >>= 1) e += __shfl_xor(e, o);
  if (j == 0) den[b] = mx2 + __logf(e);
}

// ---------------------------------------------------------------------------
// Kernel 4: CRF numerator (gold path score). One block per batch element.
// ---------------------------------------------------------------------------
__global__ __launch_bounds__(256) void crf_num_kernel(
    const float* __restrict__ logits, const int* __restrict__ mask,
    const int* __restrict__ tags, const float* __restrict__ trans,
    const float* __restrict__ startt, const float* __restrict__ endt,
    float* __restrict__ num)
{
  __shared__ float rsum[256];
  __shared__ int rlen[256];
  const int b = blockIdx.x;
  const int tid = threadIdx.x;

  float sum = 0.f;
  int len = 0;
  for (int s = tid; s < Sv; s += 256) len += mask[b * Sv + s];
  for (int s = 1 + tid; s < Sv; s += 256) {
    if (mask[b * Sv + s]) {
      int tp = tags[b * Sv + s - 1];
      int tc = tags[b * Sv + s];
      sum += trans[tp * Lv + tc] + logits[((size_t)b * Sv + s) * Lv + tc];
    }
  }
  rsum[tid] = sum;
  rlen[tid] = len;
  __syncthreads();
  for (int off = 128; off; off >>= 1) {
    if (tid < off) { rsum[tid] += rsum[tid + off]; rlen[tid] += rlen[tid + off]; }
    __syncthreads();
  }
  if (tid == 0) {
    int t0 = tags[b * Sv];
    float score = startt[t0] + logits[(size_t)b * Sv * Lv + t0] + rsum[0];
    int last = tags[b * Sv + rlen[0] - 1];
    num[b] = score + endt[last];
  }
}

// ---------------------------------------------------------------------------
// Kernel 5: loss = -mean(num - den)
// ---------------------------------------------------------------------------
__global__ __launch_bounds__(32) void crf_loss_kernel(
    const float* __restrict__ num, const float* __restrict__ den,
    float* __restrict__ out)
{
  const int j = threadIdx.x;
  float llh = num[j] - den[j];
  for (int o = 16; o; o >>= 1) llh += __shfl_xor(llh, o);
  if (j == 0) out[0] = -llh / (float)Bv;
}

extern "C" void kernel_launch(void* const* d_in, const int* in_sizes, int n_in,
                              void* d_out, int out_size, void* d_ws, size_t ws_size,
                              hipStream_t stream) {
  const float* hidden = (const float*)d_in[0];
  const int*   mask   = (const int*)d_in[1];
  const int*   labels = (const int*)d_in[2];
  const float* W      = (const float*)d_in[3];
  const float* bias   = (const float*)d_in[4];
  const float* startt = (const float*)d_in[5];
  const float* endt   = (const float*)d_in[6];
  const float* trans  = (const float*)d_in[7];

  float* out    = (float*)d_out;
  float* logits = out + 1;                    // (loss, logits) concatenated

  float* mats = (float*)d_ws;                 // B*NCH*208 floats
  float* numv = mats + (size_t)Bv * NCH * 208;
  float* denv = numv + Bv;

  // 65536 rows / 16 rows-per-wave / 8 waves-per-block = 512 blocks
  gemm_logits_kernel<<<512, 256, 0, stream>>>(hidden, W, bias, logits);
  crf_scan_kernel<<<Bv * NCH, 224, 0, stream>>>(logits, mask, trans, mats);
  crf_num_kernel<<<Bv, 256, 0, stream>>>(logits, mask, labels, trans,
                                         startt, endt, numv);
  crf_combine_kernel<<<Bv, 32, 0, stream>>>(logits, mats, startt, endt, denv);
  crf_loss_kernel<<<1, 32, 0, stream>>>(numv, denv, out);
}